// AimNet2Core_18872086298698
// MI455X (gfx1250) — compile-verified
//
#include <hip/hip_runtime.h>
#include <hip/hip_bf16.h>

#define N_ATOMS 50000
#define N_PAIRS 1000000
#define FDIM    64
#define GDIM    16
#define VDIM    32
#define H1DIM   128
#define MSG_W   132   // 129 padded to multiple of 4 (pad cols are zero)
#define DQ_OFF  (N_ATOMS * 64)
#define F_OFF   (N_ATOMS * 64 + N_ATOMS)

typedef __attribute__((ext_vector_type(2))) float v2f;
typedef __attribute__((ext_vector_type(8))) float v8f;

__device__ __forceinline__ v8f wmma4(v2f a, v2f b, v8f c) {
  // V_WMMA_F32_16X16X4_F32 : D(16x16 f32) = A(16x4 f32) x B(4x16 f32) + C
  return __builtin_amdgcn_wmma_f32_16x16x4_f32(false, a, false, b, (short)0, c,
                                               false, false);
}

__device__ __forceinline__ float gelu_exact(float x) {
  return 0.5f * x * (1.0f + erff(x * 0.70710678118654752440f));
}

// ---------------------------------------------------------------- zero ws ---
__global__ void zero_ws_kernel(float* __restrict__ w, size_t n) {
  size_t i = (size_t)blockIdx.x * blockDim.x + threadIdx.x;
  size_t stride = (size_t)gridDim.x * blockDim.x;
  for (; i < n; i += stride) w[i] = 0.0f;
}

// ------------------------------------------------- agh transpose (one-off) --
// aghT[h][a][g] = agh[a][g][h]  -> coalesced B-fragment loads in msg_kernel
__global__ void transpose_agh_kernel(const float* __restrict__ agh,
                                     float* __restrict__ aghT) {
  int i = blockIdx.x * blockDim.x + threadIdx.x;
  if (i < FDIM * GDIM * VDIM) {
    int a = i >> 9;          // / 512
    int r = i & 511;
    int g = r >> 5;          // / 32
    int h = r & 31;
    aghT[(size_t)h * (FDIM * GDIM) + a * GDIM + g] = agh[i];
  }
}

// --------------------------------------------------- pair accumulation ------
// S[j] += sum_g gs[p,g]
// M[j] += upper-triangle of gv[p]^T gv[p]  (symmetrized by the consumer)
__global__ void pair_kernel(const float* __restrict__ gs,
                            const float* __restrict__ gv,
                            const int* __restrict__ pair_idx,
                            float* __restrict__ S,
                            float* __restrict__ M) {
  int p = blockIdx.x * blockDim.x + threadIdx.x;
  if (p >= N_PAIRS) return;
  int j = pair_idx[N_PAIRS + p];  // row 1 of (2, N_PAIRS)

  const float* gsp = gs + (size_t)p * GDIM;
  float gsum = 0.f;
#pragma unroll
  for (int g = 0; g < GDIM; ++g) gsum += gsp[g];
  atomicAdd(&S[j], gsum);

  const float* gvp = gv + (size_t)p * (3 * GDIM);
  // stream prefetch for the gv read pattern (HBM-bound part of this stage)
  __builtin_prefetch(gvp + 3 * GDIM * 64, 0, 1);

  float d0[GDIM], d1[GDIM], d2[GDIM];
#pragma unroll
  for (int g = 0; g < GDIM; ++g) {
    d0[g] = gvp[g];
    d1[g] = gvp[GDIM + g];
    d2[g] = gvp[2 * GDIM + g];
  }
  float* Mj = M + (size_t)j * 256;
#pragma unroll
  for (int g = 0; g < GDIM; ++g) {
#pragma unroll
    for (int g2 = g; g2 < GDIM; ++g2) {
      float m = d0[g] * d0[g2] + d1[g] * d1[g2] + d2[g] * d2[g2];
      atomicAdd(&Mj[g * 16 + g2], m);   // upper triangle only: 137 atomics/pair
    }
  }
}

// ----------------------------------------------- per-atom message builder ---
// One wave per 16-atom tile.
//   t_tile(16 atoms x 16 g) = emb_tile(16x64) @ aghT[h]   via WMMA f32
//   vec_emb[j,h] = t_j^T M_j t_j  (per-atom quadratic form via LDS)
//   msg[:,0:64]=S*emb, msg[:,64:96]=vec_emb, msg[:,96]=S*q, rest zero.
__global__ __launch_bounds__(32) void msg_kernel(
    const float* __restrict__ emb, const float* __restrict__ q,
    const float* __restrict__ aghT, const float* __restrict__ S,
    const float* __restrict__ M, float* __restrict__ MSG) {
  __shared__ float lM[16 * 256];  // 16 KB : M matrices of the tile
  __shared__ float lT[16 * 16];   // 1 KB  : t slice for current h

  const int lane = threadIdx.x;
  const int base = blockIdx.x * 16;
  const int mrow = lane & 15;          // A-matrix row / C-matrix column
  const int koff = (lane >> 4) * 2;    // A/B K sub-offset within 4-chunk
  const int rbase = (lane >> 4) * 8;   // C-matrix row base

  // M tile -> LDS via CDNA5 async copy (ASYNCcnt path, no VGPR round-trip).
  // Generic->LDS: low 32 bits of a generic LDS address are the LDS offset.
  {
    const float* gsrc = M + (size_t)base * 256;
#pragma unroll 4
    for (int k = 0; k < 32; ++k) {
      int i = (k * 32 + lane) * 4;  // float4 granule per lane
      unsigned int ldsoff = (unsigned int)(uintptr_t)(&lM[i]);
      const float* ga = gsrc + i;
      asm volatile("global_load_async_to_lds_b128 %0, %1, off"
                   :
                   : "v"(ldsoff), "v"(ga)
                   : "memory");
    }
  }

  // Overlap with the async copy: A fragments (16 atoms x 64 feats, K chunks of 4)
  float af[16][2];
  {
    const float* arow = emb + (size_t)(base + mrow) * FDIM;
#pragma unroll
    for (int kc = 0; kc < 16; ++kc) {
      af[kc][0] = arow[kc * 4 + koff + 0];
      af[kc][1] = arow[kc * 4 + koff + 1];
    }
  }

  // radial_emb and radial_q directly into MSG (also overlaps the async copy)
  for (int i = lane; i < 16 * FDIM; i += 32) {
    int at = i >> 6, f = i & 63;
    MSG[(size_t)(base + at) * MSG_W + f] =
        S[base + at] * emb[(size_t)(base + at) * FDIM + f];
  }
  if (lane < 16)
    MSG[(size_t)(base + lane) * MSG_W + 96] = S[base + lane] * q[base + lane];

  asm volatile("s_wait_asynccnt 0x0" ::: "memory");
  __syncthreads();

  // Symmetrize: pair stage only accumulated the upper triangle (g <= g2)
  for (int i = lane; i < 16 * 256; i += 32) {
    int at = i >> 8, idx = i & 255, g = idx >> 4, g2 = idx & 15;
    if (g2 < g) lM[i] = lM[at * 256 + g2 * 16 + g];
  }
  __syncthreads();

  const int atomQ = lane >> 1;  // 2 lanes per atom for the quadratic form
  const int halfQ = lane & 1;

  for (int hh = 0; hh < VDIM; ++hh) {
    const float* bh = aghT + (size_t)hh * (FDIM * GDIM);
    v8f c = {0.f, 0.f, 0.f, 0.f, 0.f, 0.f, 0.f, 0.f};
#pragma unroll
    for (int kc = 0; kc < 16; ++kc) {
      int kb = kc * 4 + koff;
      v2f a = {af[kc][0], af[kc][1]};
      v2f b = {bh[kb * 16 + mrow], bh[(kb + 1) * 16 + mrow]};  // coalesced
      c = wmma4(a, b, c);
    }
    // C layout -> LDS: vgpr r => atom r (+8 for upper half-wave), col = mrow
#pragma unroll
    for (int r = 0; r < 8; ++r) lT[(r + rbase) * 16 + mrow] = c[r];
    __syncthreads();

    float acc = 0.f;
#pragma unroll
    for (int gg = 0; gg < 8; ++gg) {
      int g2 = halfQ * 8 + gg;
      float w = 0.f;
#pragma unroll
      for (int g = 0; g < 16; ++g)
        w += lM[atomQ * 256 + g2 * 16 + g] * lT[atomQ * 16 + g];
      acc += lT[atomQ * 16 + g2] * w;
    }
    acc += __shfl_xor(acc, 1, 32);
    if (halfQ == 0) MSG[(size_t)(base + atomQ) * MSG_W + 64 + hh] = acc;
    __syncthreads();
  }
}

// --------------------------------------------------------- fused 3-layer MLP
// One wave per 16-atom tile. h1/h2 staged through LDS to re-fragment as A.
__global__ __launch_bounds__(32) void mlp_kernel(
    const float* __restrict__ MSG, const float* __restrict__ W1,
    const float* __restrict__ b1, const float* __restrict__ W2,
    const float* __restrict__ b2, const float* __restrict__ W3,
    const float* __restrict__ b3, float* __restrict__ out) {
  __shared__ float lH[16 * H1DIM];  // 8 KB activation tile

  const int lane = threadIdx.x;
  const int base = blockIdx.x * 16;
  const int mrow = lane & 15;
  const int koff = (lane >> 4) * 2;
  const int rbase = (lane >> 4) * 8;

  // ---- layer 1: (16 x 132pad) @ W1(129x128), K guarded at 129 ----
  float a1[33][2];
  {
    const float* mr = MSG + (size_t)(base + mrow) * MSG_W;
#pragma unroll
    for (int kc = 0; kc < 33; ++kc) {
      a1[kc][0] = mr[kc * 4 + koff + 0];
      a1[kc][1] = mr[kc * 4 + koff + 1];
    }
  }
  for (int nt = 0; nt < 8; ++nt) {
    int col = nt * 16 + mrow;
    float bv = b1[col];
    v8f c = {bv, bv, bv, bv, bv, bv, bv, bv};
#pragma unroll
    for (int kc = 0; kc < 33; ++kc) {
      int kb = kc * 4 + koff;
      v2f b = {(kb < 129) ? W1[(size_t)kb * H1DIM + col] : 0.f,
               (kb + 1 < 129) ? W1[(size_t)(kb + 1) * H1DIM + col] : 0.f};
      v2f a = {a1[kc][0], a1[kc][1]};
      c = wmma4(a, b, c);
    }
#pragma unroll
    for (int r = 0; r < 8; ++r)
      lH[(r + rbase) * H1DIM + col] = gelu_exact(c[r]);
  }
  __syncthreads();

  // ---- layer 2: (16x128) @ W2(128x128) ----
  float a2[32][2];
#pragma unroll
  for (int kc = 0; kc < 32; ++kc) {
    a2[kc][0] = lH[mrow * H1DIM + kc * 4 + koff + 0];
    a2[kc][1] = lH[mrow * H1DIM + kc * 4 + koff + 1];
  }
  __syncthreads();  // all reads done before overwrite
  for (int nt = 0; nt < 8; ++nt) {
    int col = nt * 16 + mrow;
    float bv = b2[col];
    v8f c = {bv, bv, bv, bv, bv, bv, bv, bv};
#pragma unroll
    for (int kc = 0; kc < 32; ++kc) {
      int kb = kc * 4 + koff;
      v2f b = {W2[(size_t)kb * H1DIM + col], W2[(size_t)(kb + 1) * H1DIM + col]};
      v2f a = {a2[kc][0], a2[kc][1]};
      c = wmma4(a, b, c);
    }
#pragma unroll
    for (int r = 0; r < 8; ++r)
      lH[(r + rbase) * H1DIM + col] = gelu_exact(c[r]);
  }
  __syncthreads();

  // ---- layer 3: (16x128) @ W3(128x66), N guarded at 66; scatter outputs ----
  float a3[32][2];
#pragma unroll
  for (int kc = 0; kc < 32; ++kc) {
    a3[kc][0] = lH[mrow * H1DIM + kc * 4 + koff + 0];
    a3[kc][1] = lH[mrow * H1DIM + kc * 4 + koff + 1];
  }
  for (int nt = 0; nt < 5; ++nt) {
    int col = nt * 16 + mrow;
    bool vc = (col < 66);
    float bv = vc ? b3[col] : 0.f;
    v8f c = {bv, bv, bv, bv, bv, bv, bv, bv};
#pragma unroll
    for (int kc = 0; kc < 32; ++kc) {
      int kb = kc * 4 + koff;
      v2f b = {vc ? W3[(size_t)kb * 66 + col] : 0.f,
               vc ? W3[(size_t)(kb + 1) * 66 + col] : 0.f};
      v2f a = {a3[kc][0], a3[kc][1]};
      c = wmma4(a, b, c);
    }
#pragma unroll
    for (int r = 0; r < 8; ++r) {
      if (vc) {
        int at = base + r + rbase;
        float x = c[r];
        if (col == 0)
          out[DQ_OFF + at] = x;                       // delta_q
        else if (col == 1)
          out[F_OFF + at] = x;                        // f
        else
          out[(size_t)at * 64 + (col - 2)] = x;       // delta_a
      }
    }
  }
}

extern "C" void kernel_launch(void* const* d_in, const int* in_sizes, int n_in,
                              void* d_out, int out_size, void* d_ws,
                              size_t ws_size, hipStream_t stream) {
  const float* emb = (const float*)d_in[0];
  const float* q   = (const float*)d_in[1];
  const float* gs  = (const float*)d_in[2];
  const float* gv  = (const float*)d_in[3];
  const float* agh = (const float*)d_in[4];
  const float* W1  = (const float*)d_in[5];
  const float* b1  = (const float*)d_in[6];
  const float* W2  = (const float*)d_in[7];
  const float* b2  = (const float*)d_in[8];
  const float* W3  = (const float*)d_in[9];
  const float* b3  = (const float*)d_in[10];
  const int* pidx  = (const int*)d_in[11];
  float* out = (float*)d_out;

  // ws layout (floats): S[N_ATOMS] | M[N_ATOMS*256] | MSG[N_ATOMS*132] | aghT[32768]
  float* S    = (float*)d_ws;
  float* M    = S + N_ATOMS;
  float* MSG  = M + (size_t)N_ATOMS * 256;
  float* aghT = MSG + (size_t)N_ATOMS * MSG_W;
  size_t nz   = (size_t)N_ATOMS * (1 + 256 + MSG_W);

  zero_ws_kernel<<<2048, 256, 0, stream>>>((float*)d_ws, nz);
  transpose_agh_kernel<<<(FDIM * GDIM * VDIM + 255) / 256, 256, 0, stream>>>(agh,
                                                                             aghT);
  pair_kernel<<<(N_PAIRS + 255) / 256, 256, 0, stream>>>(gs, gv, pidx, S, M);
  msg_kernel<<<N_ATOMS / 16, 32, 0, stream>>>(emb, q, aghT, S, M, MSG);
  mlp_kernel<<<N_ATOMS / 16, 32, 0, stream>>>(MSG, W1, b1, W2, b2, W3, b3, out);
}